// Net_58789512348294
// MI455X (gfx1250) — compile-verified
//
#include <hip/hip_runtime.h>
#include <math.h>

#define BGR   64
#define NPG   2048
#define NEDGE 1048576
#define NTOT  131072
#define HID   128
#define NCLS  32
#define KP1   512
#define KP2   128
#define N1TOT (BGR*KP1)   /* 32768 */
#define N2TOT (BGR*KP2)   /* 8192  */

typedef __attribute__((ext_vector_type(2))) float v2f;
typedef __attribute__((ext_vector_type(8))) float v8f;

// ---------------- utility fills ----------------
__global__ void fill_f32(float* __restrict__ p, float v, int n){
  int i = blockIdx.x*blockDim.x + threadIdx.x;
  if(i<n) p[i]=v;
}
__global__ void fill_i32(int* __restrict__ p, int v, int n){
  int i = blockIdx.x*blockDim.x + threadIdx.x;
  if(i<n) p[i]=v;
}

// ---------------- degree (with self-loop pre-init 1.0) ----------------
__global__ void deg_count_raw(const int* __restrict__ dst, float* __restrict__ deg, int e){
  int i = blockIdx.x*blockDim.x + threadIdx.x;
  if(i<e) atomicAdd(&deg[dst[i]], 1.0f);
}
__global__ void deg_count_mapped(const int* __restrict__ src, const int* __restrict__ dst,
                                 const int* __restrict__ nmap, float* __restrict__ deg, int e){
  int i = blockIdx.x*blockDim.x + threadIdx.x;
  if(i>=e) return;
  int s = nmap[src[i]], d = nmap[dst[i]];
  if(s>=0 && d>=0) atomicAdd(&deg[d], 1.0f);
}

// agg[i] = t[i]/deg[i] + bias  (GCN self-loop term + bias)
__global__ void self_init(const float* __restrict__ t, const float* __restrict__ deg,
                          const float* __restrict__ bias, float* __restrict__ agg, int n){
  int i = blockIdx.x*blockDim.x + threadIdx.x;
  if(i<n) agg[i] = t[i]/deg[i] + (bias ? bias[0] : 0.0f);
}

// scalar-feature symmetric-norm edge aggregation
__global__ void edge_scalar_raw(const int* __restrict__ src, const int* __restrict__ dst,
                                const float* __restrict__ deg, const float* __restrict__ t,
                                float* __restrict__ agg, int e){
  int i = blockIdx.x*blockDim.x + threadIdx.x;
  if(i>=e) return;
  int s = src[i], d = dst[i];
  float nrm = rsqrtf(deg[s]) * rsqrtf(deg[d]);
  atomicAdd(&agg[d], t[s]*nrm);
}
__global__ void edge_scalar_mapped(const int* __restrict__ src, const int* __restrict__ dst,
                                   const int* __restrict__ nmap,
                                   const float* __restrict__ deg, const float* __restrict__ t,
                                   float* __restrict__ agg, int e){
  int i = blockIdx.x*blockDim.x + threadIdx.x;
  if(i>=e) return;
  int s = nmap[src[i]], d = nmap[dst[i]];
  if(s<0 || d<0) return;
  float nrm = rsqrtf(deg[s]) * rsqrtf(deg[d]);
  atomicAdd(&agg[d], t[s]*nrm);
}

// t1[i] = sum_f relu(aggx[i]*W1[f]+b1[f]) * sW1[f]   (h1 never materialized)
__global__ void t1_kernel(const float* __restrict__ aggx, const float* __restrict__ W1,
                          const float* __restrict__ b1, const float* __restrict__ sW1,
                          float* __restrict__ t1, int n){
  int i = blockIdx.x*blockDim.x + threadIdx.x;
  if(i>=n) return;
  float a = aggx[i], acc = 0.f;
  #pragma unroll 8
  for(int f=0; f<HID; ++f){
    float h = fmaxf(a*W1[f] + b1[f], 0.f);
    acc += h * sW1[f];
  }
  t1[i] = acc;
}

// t[i] = h[i,:] . w
__global__ void rowdot_kernel(const float* __restrict__ h, const float* __restrict__ w,
                              float* __restrict__ t, int n){
  int i = blockIdx.x*blockDim.x + threadIdx.x;
  if(i>=n) return;
  const float* row = h + (size_t)i*HID;
  float acc = 0.f;
  #pragma unroll 8
  for(int f=0; f<HID; ++f) acc += row[f]*w[f];
  t[i] = acc;
}

// ---------------- per-graph top-k via LDS bitonic sort (descending) ----------------
template<int NN, int KS>
__global__ void topk_kernel(const float* __restrict__ score, int* __restrict__ perm){
  __shared__ float ss[NN];
  __shared__ int   si[NN];
  int g = blockIdx.x;
  int tid = threadIdx.x;                 // blockDim == NN/2
  for(int i=tid; i<NN; i+=blockDim.x){ ss[i]=score[g*NN+i]; si[i]=i; }
  __syncthreads();
  for(int k=2; k<=NN; k<<=1){
    for(int j=k>>1; j>0; j>>=1){
      for(int i=tid; i<NN; i+=blockDim.x){
        int p = i ^ j;
        if(p > i){
          bool desc = ((i & k) == 0);
          float a=ss[i], b=ss[p];
          int ia=si[i], ib=si[p];
          bool aAfter = (a<b) || (a==b && ia>ib);  // 'a' ranks after 'b' in descending order
          if(desc ? aAfter : !aAfter){
            ss[i]=b; ss[p]=a; si[i]=ib; si[p]=ia;
          }
        }
      }
      __syncthreads();
    }
  }
  for(int jx=tid; jx<KS; jx+=blockDim.x)
    perm[g*KS+jx] = g*NN + si[jx];
}

__global__ void scatter_nmap(const int* __restrict__ perm, int* __restrict__ nmap, int n){
  int i = blockIdx.x*blockDim.x + threadIdx.x;
  if(i<n) nmap[perm[i]] = i;
}

// xp1[j,f] = relu(aggx[perm[j]]*W1[f]+b1[f]) * tanh(score[perm[j]])
__global__ void pool1_kernel(const int* __restrict__ perm, const float* __restrict__ aggx,
                             const float* __restrict__ score, const float* __restrict__ W1,
                             const float* __restrict__ b1, float* __restrict__ xp){
  int j = blockIdx.x, f = threadIdx.x;
  int p = perm[j];
  float h = fmaxf(aggx[p]*W1[f] + b1[f], 0.f);
  xp[(size_t)j*HID + f] = h * tanhf(score[p]);
}
__global__ void pool2_kernel(const int* __restrict__ perm, const float* __restrict__ h2,
                             const float* __restrict__ score, float* __restrict__ xp){
  int j = blockIdx.x, f = threadIdx.x;
  int p = perm[j];
  xp[(size_t)j*HID + f] = h2[(size_t)p*HID + f] * tanhf(score[p]);
}

template<int KR>
__global__ void readout_kernel(const float* __restrict__ xp, float* __restrict__ xo){
  int g = blockIdx.x, f = threadIdx.x;
  float mx = -INFINITY, sm = 0.f;
  const float* base = xp + (size_t)g*KR*HID + f;
  for(int r=0; r<KR; ++r){
    float v = base[(size_t)r*HID];
    mx = fmaxf(mx, v);
    sm += v;
  }
  xo[g*2*HID + f]       = mx;
  xo[g*2*HID + HID + f] = sm * (1.0f/KR);
}

// ---------------- C[M,128] = A[M,128] @ W[128,128], fp32 WMMA ----------------
// One wave per 16x16 C tile; K=128 chained through 32 x v_wmma_f32_16x16x4_f32.
// ISA 7.12.2 layouts: A 16x4 -> lane L holds A[L%16][k0+2*(L/16)+{0,1}];
// B 4x16 -> lane L holds B[k0+2*(L/16)+{0,1}][L%16];
// D 16x16 -> vgpr r, lane L -> D[r+8*(L/16)][L%16].
__global__ void gemm128_wmma(const float* __restrict__ A, const float* __restrict__ W,
                             float* __restrict__ C, int M){
  int gw   = (blockIdx.x*blockDim.x + threadIdx.x) >> 5;
  int lane = threadIdx.x & 31;
  const int tilesN = HID/16;
  int tiles = (M/16)*tilesN;
  if(gw >= tiles) return;                    // whole-wave uniform exit (EXEC stays all-1)
  int tm = gw / tilesN, tn = gw % tilesN;
  int half = lane >> 4, l16 = lane & 15;
  int row = tm*16 + l16;
  int col = tn*16 + l16;
  v8f acc = {};
#if defined(__gfx1250__) && __has_builtin(__builtin_amdgcn_wmma_f32_16x16x4_f32)
  for(int kk=0; kk<HID; kk+=4){
    int ka = kk + 2*half;
    v2f a, b;
    a.x = A[(size_t)row*HID + ka];
    a.y = A[(size_t)row*HID + ka + 1];
    b.x = W[(size_t)ka*HID + col];
    b.y = W[(size_t)(ka+1)*HID + col];
    acc = __builtin_amdgcn_wmma_f32_16x16x4_f32(false, a, false, b,
                                                (short)0, acc, false, false);
  }
#else
  for(int r=0; r<8; ++r){
    int m = tm*16 + r + 8*half;
    float s = 0.f;
    for(int k=0; k<HID; ++k) s += A[(size_t)m*HID+k]*W[(size_t)k*HID+col];
    acc[r] = s;
  }
#endif
  #pragma unroll
  for(int r=0; r<8; ++r){
    int m = tm*16 + r + 8*half;
    C[(size_t)m*HID + col] = acc[r];
  }
}

// h2[i,f] = hw2[i,f]/deg[i] + b2[f]   (self-loop + bias, edge adds follow, relu after)
__global__ void h2_init(const float* __restrict__ hw, const float* __restrict__ deg,
                        const float* __restrict__ b, float* __restrict__ h2, int n){
  int i = blockIdx.x*blockDim.x + threadIdx.x;
  if(i>=n) return;
  int node = i >> 7, f = i & (HID-1);
  h2[i] = hw[i]/deg[node] + b[f];
}
__global__ void relu_inplace(float* __restrict__ h, int n){
  int i = blockIdx.x*blockDim.x + threadIdx.x;
  if(i<n) h[i] = fmaxf(h[i], 0.f);
}

// 128-wide edge aggregation, one wave per edge (survivor edges only do work)
__global__ void edge_agg_feat(const int* __restrict__ src, const int* __restrict__ dst,
                              const int* __restrict__ nmap, const float* __restrict__ deg,
                              const float* __restrict__ hsrc, float* __restrict__ hacc, int e){
  int gw   = (blockIdx.x*blockDim.x + threadIdx.x) >> 5;
  int lane = threadIdx.x & 31;
  if(gw >= e) return;
  int s = nmap[src[gw]], d = nmap[dst[gw]];
  if(s<0 || d<0) return;
  float nrm = rsqrtf(deg[s]) * rsqrtf(deg[d]);
  const float* rs = hsrc + (size_t)s*HID;
  float* rd = hacc + (size_t)d*HID;
  #pragma unroll
  for(int f=lane; f<HID; f+=32)
    atomicAdd(&rd[f], rs[f]*nrm);
}

// ---------------- MLP head + log_softmax, one block per graph ----------------
__global__ void head_kernel(const float* __restrict__ x1, const float* __restrict__ x2,
                            const float* __restrict__ gi,
                            const float* __restrict__ l1W, const float* __restrict__ l1b,
                            const float* __restrict__ l2W, const float* __restrict__ l2b,
                            const float* __restrict__ l3W, const float* __restrict__ l3b,
                            float* __restrict__ out){
  __shared__ float z[2*HID+10];
  __shared__ float z1[HID];
  __shared__ float z2s[HID/2];
  __shared__ float lg[NCLS];
  __shared__ float red[2];
  int g = blockIdx.x, t = threadIdx.x;      // blockDim = 128
  for(int i=t; i<2*HID+10; i+=HID)
    z[i] = (i < 2*HID) ? (x1[g*2*HID+i] + x2[g*2*HID+i]) : gi[g*10 + (i-2*HID)];
  __syncthreads();
  {
    float acc = l1b[t];
    for(int i=0; i<2*HID+10; ++i) acc += z[i]*l1W[i*HID + t];
    z1[t] = fmaxf(acc, 0.f);
  }
  __syncthreads();
  if(t < HID/2){
    float acc = l2b[t];
    for(int i=0; i<HID; ++i) acc += z1[i]*l2W[i*(HID/2) + t];
    z2s[t] = fmaxf(acc, 0.f);
  }
  __syncthreads();
  if(t < NCLS){
    float acc = l3b[t];
    for(int i=0; i<HID/2; ++i) acc += z2s[i]*l3W[i*NCLS + t];
    lg[t] = acc;
  }
  __syncthreads();
  if(t == 0){
    float m = -INFINITY;
    for(int i=0;i<NCLS;++i) m = fmaxf(m, lg[i]);
    float s = 0.f;
    for(int i=0;i<NCLS;++i) s += expf(lg[i]-m);
    red[0] = m; red[1] = logf(s);
  }
  __syncthreads();
  if(t < NCLS) out[g*NCLS + t] = lg[t] - red[0] - red[1];
}

extern "C" void kernel_launch(void* const* d_in, const int* in_sizes, int n_in,
                              void* d_out, int out_size, void* d_ws, size_t ws_size,
                              hipStream_t stream){
  const float* x   = (const float*)d_in[0];
  const int*   src = (const int*)  d_in[1];
  const int*   dst = (const int*)  d_in[2];
  const float* gi  = (const float*)d_in[3];
  const float* W1  = (const float*)d_in[4];
  const float* b1  = (const float*)d_in[5];
  const float* sW1 = (const float*)d_in[6];
  const float* sb1 = (const float*)d_in[7];
  const float* W2  = (const float*)d_in[8];
  const float* b2  = (const float*)d_in[9];
  const float* sW2 = (const float*)d_in[10];
  const float* sb2 = (const float*)d_in[11];
  const float* l1W = (const float*)d_in[12];
  const float* l1b = (const float*)d_in[13];
  const float* l2W = (const float*)d_in[14];
  const float* l2b = (const float*)d_in[15];
  const float* l3W = (const float*)d_in[16];
  const float* l3b = (const float*)d_in[17];
  float* out = (float*)d_out;
  (void)in_sizes; (void)n_in; (void)out_size; (void)ws_size;

  size_t off = 0;
  auto alloc = [&](size_t nbytes)->void*{
    void* p = (char*)d_ws + off;
    off += (nbytes + 255) & ~(size_t)255;
    return p;
  };
  float* deg1 = (float*)alloc((size_t)NTOT*4);
  float* aggx = (float*)alloc((size_t)NTOT*4);
  float* t1   = (float*)alloc((size_t)NTOT*4);
  float* sc1  = (float*)alloc((size_t)NTOT*4);
  float* deg2 = (float*)alloc((size_t)N1TOT*4);
  float* t2   = (float*)alloc((size_t)N1TOT*4);
  float* sc2  = (float*)alloc((size_t)N1TOT*4);
  float* x1   = (float*)alloc((size_t)BGR*2*HID*4);
  float* x2   = (float*)alloc((size_t)BGR*2*HID*4);
  float* xp1  = (float*)alloc((size_t)N1TOT*HID*4);
  float* hw2  = (float*)alloc((size_t)N1TOT*HID*4);
  float* h2   = (float*)alloc((size_t)N1TOT*HID*4);
  float* xp2  = (float*)alloc((size_t)N2TOT*HID*4);
  int*   perm1= (int*)  alloc((size_t)N1TOT*4);
  int*   nmap = (int*)  alloc((size_t)NTOT*4);
  int*   perm2= (int*)  alloc((size_t)N2TOT*4);

  const int T = 256;
  // ---- stage 1: GCN folded through rank-1 weights (scalar edge work) ----
  fill_f32<<<(NTOT+T-1)/T, T, 0, stream>>>(deg1, 1.0f, NTOT);
  deg_count_raw<<<NEDGE/T, T, 0, stream>>>(dst, deg1, NEDGE);
  self_init<<<(NTOT+T-1)/T, T, 0, stream>>>(x, deg1, nullptr, aggx, NTOT);
  edge_scalar_raw<<<NEDGE/T, T, 0, stream>>>(src, dst, deg1, x, aggx, NEDGE);
  t1_kernel<<<(NTOT+T-1)/T, T, 0, stream>>>(aggx, W1, b1, sW1, t1, NTOT);
  self_init<<<(NTOT+T-1)/T, T, 0, stream>>>(t1, deg1, sb1, sc1, NTOT);
  edge_scalar_raw<<<NEDGE/T, T, 0, stream>>>(src, dst, deg1, t1, sc1, NEDGE);
  topk_kernel<NPG, KP1><<<BGR, NPG/2, 0, stream>>>(sc1, perm1);
  fill_i32<<<(NTOT+T-1)/T, T, 0, stream>>>(nmap, -1, NTOT);
  scatter_nmap<<<(N1TOT+T-1)/T, T, 0, stream>>>(perm1, nmap, N1TOT);
  pool1_kernel<<<N1TOT, HID, 0, stream>>>(perm1, aggx, sc1, W1, b1, xp1);
  readout_kernel<KP1><<<BGR, HID, 0, stream>>>(xp1, x1);
  // ---- stage 2: real 32768x128x128 GEMM on the fp32 WMMA pipe ----
  gemm128_wmma<<<(N1TOT/16)*(HID/16)/8, 256, 0, stream>>>(xp1, W2, hw2, N1TOT);
  fill_f32<<<(N1TOT+T-1)/T, T, 0, stream>>>(deg2, 1.0f, N1TOT);
  deg_count_mapped<<<NEDGE/T, T, 0, stream>>>(src, dst, nmap, deg2, NEDGE);
  h2_init<<<(N1TOT*HID)/T, T, 0, stream>>>(hw2, deg2, b2, h2, N1TOT*HID);
  edge_agg_feat<<<NEDGE*32/T, T, 0, stream>>>(src, dst, nmap, deg2, hw2, h2, NEDGE);
  relu_inplace<<<(N1TOT*HID)/T, T, 0, stream>>>(h2, N1TOT*HID);
  rowdot_kernel<<<(N1TOT+T-1)/T, T, 0, stream>>>(h2, sW2, t2, N1TOT);
  self_init<<<(N1TOT+T-1)/T, T, 0, stream>>>(t2, deg2, sb2, sc2, N1TOT);
  edge_scalar_mapped<<<NEDGE/T, T, 0, stream>>>(src, dst, nmap, deg2, t2, sc2, NEDGE);
  topk_kernel<KP1, KP2><<<BGR, KP1/2, 0, stream>>>(sc2, perm2);
  pool2_kernel<<<N2TOT, HID, 0, stream>>>(perm2, h2, sc2, xp2);
  readout_kernel<KP2><<<BGR, HID, 0, stream>>>(xp2, x2);
  // ---- head ----
  head_kernel<<<BGR, HID, 0, stream>>>(x1, x2, gi, l1W, l1b, l2W, l2b, l3W, l3b, out);
}